// WeldonModel_77352361001487
// MI455X (gfx1250) — compile-verified
//
#include <hip/hip_runtime.h>
#include <hip/hip_bf16.h>
#include <stdint.h>
#include <float.h>

// ---------------------------------------------------------------------------
// WELDON head, MI455X (gfx1250, wave32).
// Phase 1: scores = x @ w via V_WMMA_F32_16X16X4_F32, LDS-staged A tiles
//          (async global->LDS). HBM-bound: ~1.05 GB => ~45 us at 23.3 TB/s.
// Phase 2: per-(n,bin) min/max over t < L[n].
// Phase 3: sort 20 values (matches jnp.sort summation order), sum, sigmoid.
// ---------------------------------------------------------------------------

#define N_ 32
#define T_ 4000
#define D_ 2048
#define R_ 10
#define TILE_T 16
#define KC 64                           // fp32 elements of K per staged chunk
#define KG (KC / 4)                     // 16 k-groups of 4
#define CHUNKS (D_ / KC)                // 32
#define WAVES 8
#define TPB (WAVES * 32)
#define TILES_PER_N (T_ / TILE_T)       // 250 (T divisible by 16)
#define TOTAL_TILES (N_ * TILES_PER_N)  // 8000
#define GRID1 (TOTAL_TILES / WAVES)     // 1000
#define GSTRIDE 17                      // padded 16B-group stride: kills bank conflicts

typedef float v2f __attribute__((ext_vector_type(2)));
typedef float v4f __attribute__((ext_vector_type(4)));
typedef float v8f __attribute__((ext_vector_type(8)));
typedef int v4i __attribute__((ext_vector_type(4)));

#define HAVE_ASYNC_LDS 0
#if defined(__has_builtin)
#if __has_builtin(__builtin_amdgcn_global_load_async_to_lds_b128)
#undef HAVE_ASYNC_LDS
#define HAVE_ASYNC_LDS 1
#endif
#endif

typedef __attribute__((address_space(1))) v4i* g_v4i_p;   // global (AS1) int4*
typedef __attribute__((address_space(3))) v4i* l_v4i_p;   // LDS   (AS3) int4*

__device__ static inline g_v4i_p to_global_v4i(const void* p) {
  // generic global addr == AS1 addr value; integer round-trip sidesteps
  // const/addrspace conversion rules.
  return (g_v4i_p)(uintptr_t)p;
}

__device__ static inline l_v4i_p to_lds_v4i(void* p) {
  // generic shared addr = {aperture_hi32, lds_offset32}; low 32 bits are the
  // workgroup-relative LDS byte offset the hardware wants.
  return (l_v4i_p)(uintptr_t)(uint32_t)(uintptr_t)p;
}

__global__ __launch_bounds__(TPB) void weldon_scores_kernel(
    const float* __restrict__ x, const int* __restrict__ lengths,
    const float* __restrict__ w, float* __restrict__ scores) {
  __shared__ float wlds[D_];                         // 8 KB: whole w vector
  __shared__ float sbuf[WAVES][KG * GSTRIDE * 4];    // 4.25 KB A-stage per wave

  const int tid = threadIdx.x;
  for (int i = tid; i < D_ / 4; i += TPB)
    ((v4f*)wlds)[i] = ((const v4f*)w)[i];
  __syncthreads();

  const int wave = tid >> 5;
  const int lane = tid & 31;
  const int tile = blockIdx.x * WAVES + wave;
  const int n = tile / TILES_PER_N;
  const int t0 = (tile - n * TILES_PER_N) * TILE_T;
  if (t0 >= lengths[n]) return;  // timesteps >= L[n] never feed a bin

  const int h = lane >> 4;   // lane half selects K-pair {0,1} vs {2,3}
  const int ml = lane & 15;  // M row within the 16x16 tile

  float* buf = sbuf[wave];
  const float* xt = x + (size_t)(n * T_ + t0) * D_;

  v8f acc = {0.f, 0.f, 0.f, 0.f, 0.f, 0.f, 0.f, 0.f};

  for (int c = 0; c < CHUNKS; ++c) {
    const int d0 = c * KC;
    // ---- stage 16 rows x 64 cols of A into LDS; each issue moves 2 rows,
    //      half-wave reads 256B contiguous global bytes (coalesced).
#pragma unroll
    for (int i = 0; i < 8; ++i) {
      const int m = 2 * i + h;
      const int kg = ml;
      const float* gsrc = xt + (size_t)m * D_ + (d0 + kg * 4);
      float* ldst = buf + (kg * GSTRIDE + m) * 4;
#if HAVE_ASYNC_LDS
      __builtin_amdgcn_global_load_async_to_lds_b128(to_global_v4i(gsrc),
                                                     to_lds_v4i(ldst), 0, 0);
#else
      *(v4f*)ldst = *(const v4f*)gsrc;
#endif
    }
#if HAVE_ASYNC_LDS
#if __has_builtin(__builtin_amdgcn_s_wait_asynccnt)
    __builtin_amdgcn_s_wait_asynccnt(0);
#else
    asm volatile("s_wait_asynccnt 0" ::: "memory");
#endif
#endif
    // ---- 16 WMMAs consume the chunk; A/B reads hit all 64 banks exactly once
#pragma unroll
    for (int kg = 0; kg < KG; ++kg) {
      v2f a = *(const v2f*)(buf + (kg * GSTRIDE + ml) * 4 + h * 2);
      v2f b = *(const v2f*)(wlds + d0 + kg * 4 + h * 2);  // broadcast per half
      acc = __builtin_amdgcn_wmma_f32_16x16x4_f32(false, a, false, b, (short)0,
                                                  acc, false, false);
    }
  }
  // B replicated across N => every column of D equals the score; lane 0 holds
  // rows t0..t0+7 in acc[0..7], lane 16 holds rows t0+8..t0+15.
  if (ml == 0) {
    float* sp = scores + (size_t)n * T_ + t0 + h * 8;
    v4f lo = {acc[0], acc[1], acc[2], acc[3]};
    v4f hi = {acc[4], acc[5], acc[6], acc[7]};
    ((v4f*)sp)[0] = lo;
    ((v4f*)sp)[1] = hi;
  }
}

__global__ __launch_bounds__(256) void weldon_minmax_kernel(
    const float* __restrict__ scores, const int* __restrict__ lengths,
    float* __restrict__ mm) {
  __shared__ float smax[256];
  __shared__ float smin[256];
  const int n = blockIdx.x / R_;
  const int r = blockIdx.x % R_;
  const int L = lengths[n];
  const int start = (r * L) / R_;
  const int end = ((r + 1) * L + R_ - 1) / R_;  // bins may overlap by 1; never empty
  float mx = -FLT_MAX, mn = FLT_MAX;
  for (int t = start + (int)threadIdx.x; t < end; t += 256) {
    const float s = scores[(size_t)n * T_ + t];
    mx = fmaxf(mx, s);
    mn = fminf(mn, s);
  }
  smax[threadIdx.x] = mx;
  smin[threadIdx.x] = mn;
  __syncthreads();
  for (int o = 128; o > 0; o >>= 1) {
    if ((int)threadIdx.x < o) {
      smax[threadIdx.x] = fmaxf(smax[threadIdx.x], smax[threadIdx.x + o]);
      smin[threadIdx.x] = fminf(smin[threadIdx.x], smin[threadIdx.x + o]);
    }
    __syncthreads();
  }
  if (threadIdx.x == 0) {
    mm[n * 2 * R_ + r] = smax[0];        // feats layout: [maxv(10), minv(10)]
    mm[n * 2 * R_ + R_ + r] = smin[0];
  }
}

__global__ void weldon_head_kernel(const float* __restrict__ mm,
                                   float* __restrict__ out) {
  const int n = threadIdx.x;
  if (n >= N_) return;
  float f[2 * R_];
#pragma unroll
  for (int i = 0; i < 2 * R_; ++i) f[i] = mm[n * 2 * R_ + i];
  for (int i = 1; i < 2 * R_; ++i) {  // ascending insertion sort == jnp.sort
    const float v = f[i];
    int j = i - 1;
    for (; j >= 0 && f[j] > v; --j) f[j + 1] = f[j];
    f[j + 1] = v;
  }
  float s = 0.f;
  for (int i = 0; i < 2 * R_; ++i) s += f[i];  // sum in sorted order
  out[n] = 1.f / (1.f + expf(-s));
}

extern "C" void kernel_launch(void* const* d_in, const int* in_sizes, int n_in,
                              void* d_out, int out_size, void* d_ws, size_t ws_size,
                              hipStream_t stream) {
  const float* x = (const float*)d_in[0];
  const int* lengths = (const int*)d_in[1];
  const float* w = (const float*)d_in[2];

  float* scores = (float*)d_ws;            // N*T floats (512 KB)
  float* mm = scores + (size_t)N_ * T_;    // N*2R floats

  weldon_scores_kernel<<<GRID1, TPB, 0, stream>>>(x, lengths, w, scores);
  weldon_minmax_kernel<<<N_ * R_, 256, 0, stream>>>(scores, lengths, mm);
  weldon_head_kernel<<<1, 32, 0, stream>>>(mm, (float*)d_out);

  (void)in_sizes; (void)n_in; (void)out_size; (void)ws_size;
}